// MultisliceForwardModel_45406394253341
// MI455X (gfx1250) — compile-verified
//
#include <hip/hip_runtime.h>

typedef float v2f __attribute__((ext_vector_type(2)));
typedef float v8f __attribute__((ext_vector_type(8)));

#define PH      128
#define NPIX    (PH * PH)      // 16384
#define NSLICES 8
#define NMODES  4
#define OBJW    512
#define NTHREADS 512           // 16 wave32 waves

// LDS partition (floats)
#define OFF_EXR   0
#define OFF_EXI   16384
#define OFF_TMR   32768
#define OFF_TMI   49152
#define OFF_WTAB  65536        // 128 x (cos,sin): w[t] = exp(-2*pi*i*t/128)
#define OFF_PTAB  65792        // 128 x (cos,sin): p[t]/128 (k-space propagator)
#define OFF_GTAB  66048        // 128 x (re,im):  g = DFT(p/128)  (real-space kernel)
#define SMEM_FLOATS 66304      // 265216 bytes (<= 320KB WGP LDS)

#define PI_F 3.14159265358979323846f

// Complex GEMM on the f32 WMMA path, one matrix synthesized from a 128-entry
// LDS table `tab` (interleaved re,im):
//   CIRC==0 (DFT):        T[r,c] = tab[(r*c)   & 127]
//   CIRC==1 (circulant):  LEFT: T[j,k] = tab[(k-j) & 127]
//                         RIGHT:T[k,n] = tab[(k-n) & 127]
//   LEFT==1 : D = T * S        LEFT==0 : D = S * T
// S, D: 128x128 row-major complex (split planes) in LDS.
// Tiling: 8x8 grid of 16x16 tiles; each of the 16 waves owns 4 tiles that
// share the table-side operand (same tile-row if LEFT, same tile-col if RIGHT),
// so the table gather feeds 16 WMMAs per k-step.
// EPI (RIGHT only): 0 = plain store, 1 = multiply by object patch then store,
//                   2 = fftshift + |.|^2 accumulate to global (no LDS store).
template <int LEFT, int CIRC, int EPI>
__device__ __forceinline__ void cgemm(const float* __restrict__ sR,
                                      const float* __restrict__ sI,
                                      float* __restrict__ dR,
                                      float* __restrict__ dI,
                                      const float* __restrict__ tab,
                                      const float* __restrict__ eR,  // EPI=1: obj_re slice base
                                      const float* __restrict__ eI,  // EPI=1: obj_im slice base
                                      int py, int px,
                                      float* __restrict__ gout,      // EPI=2: out + n*NPIX
                                      int m) {
  const int lane   = threadIdx.x & 31;
  const int wv     = threadIdx.x >> 5;   // 0..15
  const int lhalf  = lane >> 4;          // A/B: K pair select; C/D: M +0/+8
  const int l15    = lane & 15;
  const int shared = (wv >> 1) * 16;     // shared tile row (LEFT) / col (RIGHT)
  const int grp    = (wv & 1) * 4;       // first of this wave's 4 tiles

  const v8f vzero = {0.f, 0.f, 0.f, 0.f, 0.f, 0.f, 0.f, 0.f};
  v8f cr[4], ci[4];
#pragma unroll
  for (int i = 0; i < 4; ++i) { cr[i] = vzero; ci[i] = vzero; }

  for (int kb = 0; kb < 128; kb += 4) {
    const int k0 = kb + lhalf * 2;
    const int k1 = k0 + 1;

    // Table-side operand, shared by all 4 tiles of this wave.
    v2f tR, tI;
    {
      const int rc = shared + l15;       // row j (LEFT) or col n (RIGHT)
      const int i0 = (CIRC ? (LEFT ? (k0 - rc) : (k0 - rc)) : (rc * k0)) & 127;
      const int i1 = (CIRC ? (LEFT ? (k1 - rc) : (k1 - rc)) : (rc * k1)) & 127;
      tR.x = tab[2 * i0]; tI.x = tab[2 * i0 + 1];
      tR.y = tab[2 * i1]; tI.y = tab[2 * i1 + 1];
    }
    v2f tIn; tIn.x = -tI.x; tIn.y = -tI.y;

#pragma unroll
    for (int i = 0; i < 4; ++i) {
      if (LEFT) {
        // B = S tile (4x16): col n, k pair per lane half (strided rows)
        const int n = (grp + i) * 16 + l15;
        v2f bR, bI;
        bR.x = sR[k0 * PH + n];  bR.y = sR[k1 * PH + n];
        bI.x = sI[k0 * PH + n];  bI.y = sI[k1 * PH + n];
        // (tR + i tI)(bR + i bI)
        cr[i] = __builtin_amdgcn_wmma_f32_16x16x4_f32(false, tR,  false, bR, (short)0, cr[i], false, false);
        cr[i] = __builtin_amdgcn_wmma_f32_16x16x4_f32(false, tIn, false, bI, (short)0, cr[i], false, false);
        ci[i] = __builtin_amdgcn_wmma_f32_16x16x4_f32(false, tR,  false, bI, (short)0, ci[i], false, false);
        ci[i] = __builtin_amdgcn_wmma_f32_16x16x4_f32(false, tI,  false, bR, (short)0, ci[i], false, false);
      } else {
        // A = S tile (16x4): row j, contiguous k pair
        const int j = (grp + i) * 16 + l15;
        v2f aR, aI;
        aR.x = sR[j * PH + k0];  aR.y = sR[j * PH + k1];
        aI.x = sI[j * PH + k0];  aI.y = sI[j * PH + k1];
        // (aR + i aI)(tR + i tI)
        cr[i] = __builtin_amdgcn_wmma_f32_16x16x4_f32(false, aR, false, tR,  (short)0, cr[i], false, false);
        cr[i] = __builtin_amdgcn_wmma_f32_16x16x4_f32(false, aI, false, tIn, (short)0, cr[i], false, false);
        ci[i] = __builtin_amdgcn_wmma_f32_16x16x4_f32(false, aR, false, tI,  (short)0, ci[i], false, false);
        ci[i] = __builtin_amdgcn_wmma_f32_16x16x4_f32(false, aI, false, tR,  (short)0, ci[i], false, false);
      }
    }
  }

  // Store / epilogue. C/D layout: VGPR v -> row tm + v + 8*lhalf, col tn + l15.
#pragma unroll
  for (int i = 0; i < 4; ++i) {
    const int tm = LEFT ? shared : (grp + i) * 16;
    const int tn = LEFT ? (grp + i) * 16 : shared;
#pragma unroll
    for (int v = 0; v < 8; ++v) {
      const int row = tm + v + lhalf * 8;
      const int col = tn + l15;
      const float re = cr[i][v];
      const float im = ci[i][v];
      if (EPI == 2) {
        const float val = re * re + im * im;
        const int oidx = (((row + 64) & 127) << 7) | ((col + 64) & 127);
        if (m == 0) gout[oidx] = val;
        else        gout[oidx] += val;
      } else if (EPI == 1) {
        const int go = (py + row) * OBJW + (px + col);
        const float orr = eR[go], oii = eI[go];
        dR[row * PH + col] = re * orr - im * oii;
        dI[row * PH + col] = re * oii + im * orr;
      } else {
        dR[row * PH + col] = re;
        dI[row * PH + col] = im;
      }
    }
  }
}

__global__ void __launch_bounds__(NTHREADS)
multislice_fwd_kernel(const float* __restrict__ probe_r,
                      const float* __restrict__ probe_i,
                      const float* __restrict__ obj_r,
                      const float* __restrict__ obj_i,
                      const int* __restrict__ positions,
                      float* __restrict__ out) {
  extern __shared__ float smem[];
  float* exr  = smem + OFF_EXR;
  float* exi  = smem + OFF_EXI;
  float* tmr  = smem + OFF_TMR;
  float* tmi  = smem + OFF_TMI;
  float* wtab = smem + OFF_WTAB;
  float* ptab = smem + OFF_PTAB;
  float* gtab = smem + OFF_GTAB;

  const int tid = threadIdx.x;
  const int n   = blockIdx.x;
  const int py  = positions[2 * n + 0];
  const int px  = positions[2 * n + 1];
  float* gout = out + (size_t)n * NPIX;

  // Tables: twiddles w[t]; k-space propagator p[t]/128 (ifft scale folded, half
  // per side); then the fused real-space circulant kernel g = DFT(p/128).
  if (tid < PH) {
    const float ang = -2.0f * PI_F * (float)tid / 128.0f;
    wtab[2 * tid]     = cosf(ang);
    wtab[2 * tid + 1] = sinf(ang);
    const float f  = ((tid < 64) ? (float)tid : (float)(tid - 128)) / (128.0f * 0.2f);
    const float ph = (-PI_F * 0.025f * 2.0f) * f * f;
    ptab[2 * tid]     = cosf(ph) * (1.0f / 128.0f);
    ptab[2 * tid + 1] = sinf(ph) * (1.0f / 128.0f);
  }
  __syncthreads();
  if (tid < PH) {
    float gr = 0.f, gi = 0.f;
    for (int t = 0; t < PH; ++t) {
      const int iw = (t * tid) & 127;
      const float wc = wtab[2 * iw], ws = wtab[2 * iw + 1];
      const float pc = ptab[2 * t],  ps = ptab[2 * t + 1];
      gr += pc * wc - ps * ws;
      gi += pc * ws + ps * wc;
    }
    gtab[2 * tid]     = gr;
    gtab[2 * tid + 1] = gi;
  }
  __syncthreads();

  for (int m = 0; m < NMODES; ++m) {
    // ex = probe[m] * patch0
#pragma unroll 4
    for (int i = 0; i < NPIX / NTHREADS; ++i) {
      const int e = i * NTHREADS + tid;
      const int y = e >> 7, x = e & 127;
      const float pr = probe_r[m * NPIX + e];
      const float pi = probe_i[m * NPIX + e];
      const int  go = (py + y) * OBJW + (px + x);
      const float orr = obj_r[go], oii = obj_i[go];
      exr[e] = pr * orr - pi * oii;
      exi[e] = pr * oii + pi * orr;
    }
    __syncthreads();

    for (int s = 1; s < NSLICES; ++s) {
      const float* objRs = obj_r + (size_t)s * OBJW * OBJW;
      const float* objIs = obj_i + (size_t)s * OBJW * OBJW;

      // Pull this slice's patch toward the caches while the GEMMs run.
      if (tid < PH) {
        const char* prow = (const char*)(objRs + (py + tid) * OBJW + px);
        const char* irow = (const char*)(objIs + (py + tid) * OBJW + px);
        __builtin_prefetch(prow, 0, 1);  __builtin_prefetch(prow + 256, 0, 1);
        __builtin_prefetch(irow, 0, 1);  __builtin_prefetch(irow + 256, 0, 1);
      }

      // Fused fft2 -> xprop -> ifft2 as circulant pair; slice transmission
      // folded into the right-GEMM store epilogue.
      cgemm<1, 1, 0>(exr, exi, tmr, tmi, gtab, nullptr, nullptr, 0, 0, nullptr, 0);
      __syncthreads();
      cgemm<0, 1, 1>(tmr, tmi, exr, exi, gtab, objRs, objIs, py, px, nullptr, 0);
      __syncthreads();
    }

    // Final fft2; fftshift + |.|^2 mode accumulation fused into the epilogue.
    cgemm<1, 0, 0>(exr, exi, tmr, tmi, wtab, nullptr, nullptr, 0, 0, nullptr, 0);
    __syncthreads();
    cgemm<0, 0, 2>(tmr, tmi, exr, exi, wtab, nullptr, nullptr, 0, 0, gout, m);
    __syncthreads();
  }
}

extern "C" void kernel_launch(void* const* d_in, const int* in_sizes, int n_in,
                              void* d_out, int out_size, void* d_ws, size_t ws_size,
                              hipStream_t stream) {
  (void)in_sizes; (void)n_in; (void)out_size; (void)d_ws; (void)ws_size;
  const float* probe_r   = (const float*)d_in[0];
  const float* probe_i   = (const float*)d_in[1];
  const float* obj_r     = (const float*)d_in[2];
  const float* obj_i     = (const float*)d_in[3];
  const int*   positions = (const int*)d_in[4];
  float* out = (float*)d_out;

  const size_t shmem = SMEM_FLOATS * sizeof(float);
  multislice_fwd_kernel<<<dim3(512), dim3(NTHREADS), shmem, stream>>>(
      probe_r, probe_i, obj_r, obj_i, positions, out);
}